// ConvolutionalFilterManifold_82738249990778
// MI455X (gfx1250) — compile-verified
//
#include <hip/hip_runtime.h>
#include <stdint.h>

typedef __attribute__((ext_vector_type(2))) float v2f;
typedef __attribute__((ext_vector_type(8))) float v8f;

#define BATCH 32
#define IN_C  8
#define OUT_C 8
#define HW    512
#define MC    32

// ---------------- kernel 1: manifold -> per-sample filters + bias ----------------
__global__ void cfm_manifold(const float* __restrict__ q,
                             const float* __restrict__ wm1, const float* __restrict__ bm1,
                             const float* __restrict__ wm2, const float* __restrict__ bm2,
                             const float* __restrict__ wt,  const float* __restrict__ bt,
                             const float* __restrict__ wb,  const float* __restrict__ bb,
                             float* __restrict__ wfilt, float* __restrict__ bias)
{
    __shared__ float m0[MC];
    __shared__ float m1[MC];
    const int b = blockIdx.x;
    const int t = threadIdx.x;

    if (t < MC) {
        float s = bm1[t];
        const float* qb = q + b * 64;
        const float* wr = wm1 + t * 64;
        #pragma unroll
        for (int j = 0; j < 64; ++j) s += qb[j] * wr[j];
        m0[t] = s > 0.f ? s : 0.01f * s;   // leaky_relu(0.01)
    }
    __syncthreads();
    if (t < MC) {
        float s = bm2[t];
        #pragma unroll
        for (int c = 0; c < MC; ++c) s += m0[c] * wm2[t * MC + c];
        m1[t] = s > 0.f ? s : 0.01f * s;
    }
    __syncthreads();

    // wfilt[b][oc][ic][kh][kw], 576 floats per sample
    for (int o = t; o < OUT_C * IN_C * 9; o += blockDim.x) {
        float s = bt[o / 9];
        #pragma unroll
        for (int c = 0; c < MC; ++c) s += m1[c] * wt[c * (OUT_C * IN_C * 9) + o];
        wfilt[b * (OUT_C * IN_C * 9) + o] = s;
    }
    if (t < OUT_C) {
        float s = bb[t];
        #pragma unroll
        for (int c = 0; c < MC; ++c) s += m1[c] * wb[t * MC + c];
        bias[b * OUT_C + t] = s;
    }
}

// ---------------- kernel 2: per-sample conv via f32 WMMA ----------------
#define TR 16              // output rows per block
#define TW 64              // output cols per block
#define LR 18              // tile rows with halo
#define LC 66              // tile cols with halo
#define LP 10              // padded ic pitch (bank-conflict-free, keeps 8B align)
#define TILE_ELEMS (LR * LC * LP)

__global__ void __launch_bounds__(256)
cfm_conv(const float* __restrict__ x,
         const float* __restrict__ wfilt,
         const float* __restrict__ bias,
         float* __restrict__ y)
{
    __shared__ float tile[TILE_ELEMS + 16];   // +16 dummy slots for OOB async lanes

    const int b   = blockIdx.z;
    const int h0  = blockIdx.y * TR;
    const int w0  = blockIdx.x * TW;
    const int tid = threadIdx.x;

    const unsigned lds_base = (unsigned)(uintptr_t)(&tile[0]);
    const unsigned long long xbase = (unsigned long long)(uintptr_t)x;

    // ---- stage halo'd x tile into LDS via async global->LDS copies ----
    for (int i = tid; i < IN_C * LR * LC; i += 256) {
        int ic = i / (LR * LC);
        int r2 = i - ic * (LR * LC);
        int rr = r2 / LC;
        int cc = r2 - rr * LC;
        int gh = h0 - 1 + rr;
        int gw = w0 - 1 + cc;
        bool valid = (gh >= 0) && (gh < HW) && (gw >= 0) && (gw < HW);
        int ghc = gh < 0 ? 0 : (gh > HW - 1 ? HW - 1 : gh);
        int gwc = gw < 0 ? 0 : (gw > HW - 1 ? HW - 1 : gw);
        unsigned ldst = lds_base + 4u * (valid ? (unsigned)((rr * LC + cc) * LP + ic)
                                               : (unsigned)(TILE_ELEMS + (tid & 15)));
        unsigned goff = 4u * ((((unsigned)(b * IN_C + ic) * HW + (unsigned)ghc) * HW)
                              + (unsigned)gwc);
        asm volatile("global_load_async_to_lds_b32 %0, %1, %2"
                     :: "v"(ldst), "v"(goff), "s"(xbase) : "memory");
        if (!valid) tile[(rr * LC + cc) * LP + ic] = 0.f;   // disjoint from async dests
    }
    asm volatile("s_wait_asynccnt 0x0" ::: "memory");
    __syncthreads();

    const int lane = tid & 31;
    const int hi   = lane >> 4;     // lane half selects K rows 2hi,2hi+1
    const int nn   = lane & 15;     // N index = pixel within 16-wide segment
    const int wv   = tid >> 5;      // wave id (8 waves)

    // ---- per-wave A fragments: weights, A[m=oc][k=ic], 9 taps x 2 K-chunks ----
    const float* wf = wfilt + b * (OUT_C * IN_C * 9);
    v2f afrag[9][2];
    #pragma unroll
    for (int tap = 0; tap < 9; ++tap) {
        #pragma unroll
        for (int g = 0; g < 2; ++g) {
            int k0 = 4 * g + 2 * hi;
            float a0 = 0.f, a1 = 0.f;
            if (nn < OUT_C) {
                a0 = wf[nn * (IN_C * 9) + k0 * 9 + tap];
                a1 = wf[nn * (IN_C * 9) + (k0 + 1) * 9 + tap];
            }
            afrag[tap][g].x = a0;
            afrag[tap][g].y = a1;
        }
    }

    // ---- bias folded into accumulator init: D vgpr j holds row m = j + 8*hi ----
    float binit[8];
    #pragma unroll
    for (int j = 0; j < 8; ++j)
        binit[j] = (hi == 0) ? bias[b * OUT_C + j] : 0.f;

    // ---- 8 C-tiles per wave: rows {2*wv, 2*wv+1}, 4 segments of 16 pixels ----
    for (int t = 0; t < 8; ++t) {
        int rloc = wv * 2 + (t >> 2);
        int seg  = t & 3;

        v8f c;
        #pragma unroll
        for (int j = 0; j < 8; ++j) c[j] = binit[j];

        #pragma unroll
        for (int kh = 0; kh < 3; ++kh) {
            int rr = rloc + kh;
            #pragma unroll
            for (int kw = 0; kw < 3; ++kw) {
                int cc = seg * 16 + kw + nn;
                const float* p = &tile[(rr * LC + cc) * LP + 2 * hi];
                v2f b0 = *(const v2f*)(p);       // ic = 2hi, 2hi+1     (K-chunk 0)
                v2f b1 = *(const v2f*)(p + 4);   // ic = 4+2hi, 4+2hi+1 (K-chunk 1)
                c = __builtin_amdgcn_wmma_f32_16x16x4_f32(
                        false, afrag[kh * 3 + kw][0], false, b0, (short)0, c, false, false);
                c = __builtin_amdgcn_wmma_f32_16x16x4_f32(
                        false, afrag[kh * 3 + kw][1], false, b1, (short)0, c, false, false);
            }
        }

        // lanes 0-15 hold oc rows 0..7 in c[0..7]; lane = pixel -> coalesced stores
        if (hi == 0) {
            int gr = h0 + rloc;
            int gc = w0 + seg * 16 + nn;
            #pragma unroll
            for (int j = 0; j < 8; ++j)
                y[(((b * OUT_C + j) * HW) + gr) * HW + gc] = c[j];
        }
    }
}

// ---------------- launcher ----------------
extern "C" void kernel_launch(void* const* d_in, const int* in_sizes, int n_in,
                              void* d_out, int out_size, void* d_ws, size_t ws_size,
                              hipStream_t stream) {
    const float* q   = (const float*)d_in[0];
    const float* x   = (const float*)d_in[1];
    const float* wm1 = (const float*)d_in[2];
    const float* bm1 = (const float*)d_in[3];
    const float* wm2 = (const float*)d_in[4];
    const float* bm2 = (const float*)d_in[5];
    const float* wt  = (const float*)d_in[6];
    const float* bt  = (const float*)d_in[7];
    const float* wb  = (const float*)d_in[8];
    const float* bb  = (const float*)d_in[9];
    float* out = (float*)d_out;

    float* wfilt = (float*)d_ws;                         // 32*576 floats
    float* bias  = wfilt + BATCH * OUT_C * IN_C * 9;     // 32*8 floats

    cfm_manifold<<<BATCH, 64, 0, stream>>>(q, wm1, bm1, wm2, bm2, wt, bt, wb, bb,
                                           wfilt, bias);
    cfm_conv<<<dim3(HW / TW, HW / TR, BATCH), 256, 0, stream>>>(x, wfilt, bias, out);
}